// MpnCovResNet_37434934952799
// MI455X (gfx1250) — compile-verified
//
#include <hip/hip_runtime.h>
#include <hip/hip_bf16.h>

// ---------------------------------------------------------------------------
// MPN-COV head on gfx1250.
//  - All matmuls on V_WMMA_F32_16X16X4_F32 (native fp32 matrix path).
//  - All three GEMM families (covariance SYRK, 30x Newton-Schulz, FC) stage
//    tiles with the Tensor Data Mover, double-buffered against WMMA and
//    synced with s_wait_tensorcnt + workgroup split barrier.
//  - Every WMMA fragment is a single aligned ds_load_b64: A tiles are [m][k]
//    k-contiguous (pitch 18, bank-conflict-free), and B tiles use the fact
//    that every NS matrix is symmetric (polynomials in the symmetric Y0) /
//    covariance is a SYRK / FC weight is [class][k], so B is loaded [n][k].
// ---------------------------------------------------------------------------

typedef __attribute__((ext_vector_type(2))) float v2f;
typedef __attribute__((ext_vector_type(8))) float v8f;
typedef __attribute__((ext_vector_type(4))) unsigned int v4u;
typedef __attribute__((ext_vector_type(4))) int v4i;
typedef __attribute__((ext_vector_type(8))) int v8i;

#define DD   256           // D
#define NN   196           // N
#define BB   256           // batch
#define CLS  1000
#define COVO 32896         // D*(D+1)/2
#define TK   16            // K-tile for all TDM GEMMs
#define PT   18            // [row][k] LDS pitch: 16 + 2 pad (even, conflict-free)

// D = A(16x4) * B(4x16) + C   -- fp32 WMMA
__device__ __forceinline__ v8f wmma4(v2f a, v2f b, v8f c) {
  return __builtin_amdgcn_wmma_f32_16x16x4_f32(false, a, false, b, (short)0, c,
                                               false, false);
}

// ---------------------------------------------------------------------------
// Tensor Data Mover: 2D tile (tile_d0 dwords contiguous x tile_d1 rows) from
// global to LDS with row padding to pitch PT.  D# per CDNA5 ISA ch.8.
// tensor_d0/tensor_d1 clip the tile (OOB elements land as zeros in LDS).
__device__ __forceinline__ void tdm_load_2d(unsigned lds_off, const void* gptr,
                                            unsigned tile_d0, unsigned tile_d1,
                                            unsigned tensor_d0,
                                            unsigned tensor_d1,
                                            unsigned long long row_stride) {
  const unsigned long long ga = (unsigned long long)(uintptr_t)gptr;
  v4u g0;
  g0.x = 1u;  // count=1, is_restore=0, gather off
  g0.y = lds_off;
  g0.z = (unsigned)(ga & 0xffffffffu);
  g0.w = (unsigned)((ga >> 32) & 0x1ffffffu) | (2u << 30);  // type=2 (image)
  v8i g1;
  g1[0] = (int)((2u << 16)        // data_size = 4 bytes
                | (1u << 20)      // pad_enable
                | (3u << 22)      // pad_interval: every 16 dwords
                | (1u << 25));    // pad_amount: 2 dwords -> pitch 18
  g1[1] = (int)((tensor_d0 & 0xffffu) << 16);
  g1[2] = (int)(((tensor_d0 >> 16) & 0xffffu) | ((tensor_d1 & 0xffffu) << 16));
  g1[3] = (int)(((tensor_d1 >> 16) & 0xffffu) | ((tile_d0 & 0xffffu) << 16));
  g1[4] = (int)(tile_d1 & 0xffffu);                       // tile_dim1
  g1[5] = (int)(unsigned)(row_stride & 0xffffffffu);      // dim0_stride lo
  g1[6] = (int)(unsigned)((row_stride >> 32) & 0xffffu);  // dim0_stride hi
  g1[7] = 0;
  v4i gz = {0, 0, 0, 0};
#if defined(__clang_major__) && __clang_major__ >= 23
  v8i gz8 = {0, 0, 0, 0, 0, 0, 0, 0};
  __builtin_amdgcn_tensor_load_to_lds(g0, g1, gz, gz, gz8, 0);
#else
  __builtin_amdgcn_tensor_load_to_lds(g0, g1, gz, gz, 0);
#endif
}

__device__ __forceinline__ unsigned lds_offset_of(const void* p) {
  return (unsigned)(unsigned long long)(uintptr_t)p;  // low 32b = LDS offset
}

// ---------------------------------------------------------------------------
// Wave-level 64x32 MMA; A [m][k] and B [n][k], both pitch PT, all b64 loads.
__device__ __forceinline__ void mma_tile(const float* la, const float* lb,
                                         int m0, int n0, v8f acc[4][2]) {
  const int lane = threadIdx.x & 31;
  const int l15 = lane & 15;
  const int koff = (lane >> 4) << 1;
#pragma unroll
  for (int kk = 0; kk < TK; kk += 4) {
    const int k = kk + koff;
    v2f bf[2];
#pragma unroll
    for (int j = 0; j < 2; ++j)
      bf[j] = *(const v2f*)(lb + (n0 + j * 16 + l15) * PT + k);
#pragma unroll
    for (int i = 0; i < 4; ++i) {
      const v2f af = *(const v2f*)(la + (m0 + i * 16 + l15) * PT + k);
#pragma unroll
      for (int j = 0; j < 2; ++j) acc[i][j] = wmma4(af, bf[j], acc[i][j]);
    }
  }
}

// ---------------------------------------------------------------------------
// Fused mean + centering: one wave per (b,d) row; Ac = x - mean(x).
__global__ __launch_bounds__(256) void center_kernel(const float* __restrict__ x,
                                                     float* __restrict__ Ac) {
  const int warp = threadIdx.x >> 5, lane = threadIdx.x & 31;
  const int row = blockIdx.x * 8 + warp;  // < B*D
  const float* xr = x + (size_t)row * NN;
  float* ar = Ac + (size_t)row * NN;
  float s = 0.f;
  for (int n = lane; n < NN; n += 32) s += xr[n];
#pragma unroll
  for (int o = 16; o; o >>= 1) s += __shfl_xor(s, o, 32);  // all lanes get sum
  const float m = s * (1.0f / (float)NN);
  for (int n = lane; n < NN; n += 32) ar[n] = xr[n] - m;
}

// ---------------------------------------------------------------------------
// P[b] = Ac Ac^T / N  (SYRK: both TDM tiles are rows of Ac; K=196 ragged tail
// handled by TDM OOB clipping -> zero fill).
__global__ __launch_bounds__(256) void cov_gemm(const float* __restrict__ Ac,
                                                float* __restrict__ P) {
  __shared__ float la[2][128 * PT];
  __shared__ float lb[2][128 * PT];
  const int b = blockIdx.y;
  const float* Ab = Ac + (size_t)b * DD * NN;
  float* Pb = P + (size_t)b * DD * DD;
  const int rowBase = (blockIdx.x & 1) * 128;
  const int colBase = (blockIdx.x >> 1) * 128;
  const int tid = threadIdx.x;
  const int wave = tid >> 5;
  const int m0 = (wave >> 2) * 64;
  const int n0 = (wave & 3) * 32;
  v8f acc[4][2];
#pragma unroll
  for (int i = 0; i < 4; ++i)
#pragma unroll
    for (int j = 0; j < 2; ++j) acc[i][j] = (v8f){0.f};

  const bool issuer = (wave == 0);
  const int NT = (NN + TK - 1) / TK;  // 13, last tile clipped to 4 valid k
#define COV_ISSUE(t)                                                           \
  do {                                                                         \
    const int kb_ = (t) * TK;                                                  \
    const int bi_ = (t) & 1;                                                   \
    tdm_load_2d(lds_offset_of(&la[bi_][0]), Ab + rowBase * NN + kb_, TK, 128,  \
                (unsigned)(NN - kb_), 128, NN);                                \
    tdm_load_2d(lds_offset_of(&lb[bi_][0]), Ab + colBase * NN + kb_, TK, 128,  \
                (unsigned)(NN - kb_), 128, NN);                                \
  } while (0)

  if (issuer) COV_ISSUE(0);
  for (int t = 0; t < NT; ++t) {
    if (issuer) {
      if (t + 1 < NT) {
        COV_ISSUE(t + 1);
        __builtin_amdgcn_s_wait_tensorcnt(2);
      } else {
        __builtin_amdgcn_s_wait_tensorcnt(0);
      }
    }
    __syncthreads();
    mma_tile(&la[t & 1][0], &lb[t & 1][0], m0, n0, acc);
    __syncthreads();
  }
#undef COV_ISSUE

  const int lane = tid & 31, l15 = lane & 15, rh = (lane >> 4) * 8;
#pragma unroll
  for (int i = 0; i < 4; ++i)
#pragma unroll
    for (int j = 0; j < 2; ++j)
#pragma unroll
      for (int r = 0; r < 8; ++r) {
        const int row = rowBase + m0 + i * 16 + rh + r;
        const int col = colBase + n0 + j * 16 + l15;
        Pb[row * DD + col] = acc[i][j][r] * (1.0f / (float)NN);
      }
}

// ---------------------------------------------------------------------------
__global__ __launch_bounds__(256) void norm_kernel(const float* __restrict__ P,
                                                   float* __restrict__ norm) {
  __shared__ float red[256];
  const int b = blockIdx.x;
  const float* Pb = P + (size_t)b * DD * DD;
  float s = 0.f;
  for (int i = threadIdx.x; i < DD * DD; i += 256) {
    const float v = Pb[i];
    s += v * v;
  }
  red[threadIdx.x] = s;
  __syncthreads();
  for (int st = 128; st; st >>= 1) {
    if (threadIdx.x < st) red[threadIdx.x] += red[threadIdx.x + st];
    __syncthreads();
  }
  if (threadIdx.x == 0) norm[b] = sqrtf(red[0]);
}

__global__ __launch_bounds__(256) void prep_kernel(float* __restrict__ Y,
                                                   float* __restrict__ Z,
                                                   const float* __restrict__ norm) {
  const size_t gid = (size_t)blockIdx.x * 256 + threadIdx.x;
  const int b = (int)(gid >> 16);
  const int ij = (int)(gid & 65535);
  Y[gid] *= 1.0f / norm[b];
  Z[gid] = ((ij >> 8) == (ij & 255)) ? 1.0f : 0.0f;
}

// ---------------------------------------------------------------------------
// C[b] = scale * A[b] @ Bm[b] + diag*I.  All NS matrices are symmetric, so the
// B tile is loaded transposed-for-free as rows [n][k] (B[k][n] == B[n][k]).
__global__ __launch_bounds__(256) void ns_gemm(const float* __restrict__ A,
                                               const float* __restrict__ Bm,
                                               float* __restrict__ C,
                                               float scale, float diag) {
  __shared__ float la[2][128 * PT];
  __shared__ float lb[2][128 * PT];
  const size_t mat = (size_t)blockIdx.y * DD * DD;
  A += mat; Bm += mat; C += mat;
  const int rowBase = (blockIdx.x & 1) * 128;
  const int colBase = (blockIdx.x >> 1) * 128;
  const int tid = threadIdx.x;
  const int wave = tid >> 5;
  const int m0 = (wave >> 2) * 64;
  const int n0 = (wave & 3) * 32;
  v8f acc[4][2];
#pragma unroll
  for (int i = 0; i < 4; ++i)
#pragma unroll
    for (int j = 0; j < 2; ++j) acc[i][j] = (v8f){0.f};

  const bool issuer = (wave == 0);
  const int NT = DD / TK;  // 16 k-tiles
#define NS_ISSUE(t)                                                            \
  do {                                                                         \
    const int kb_ = (t) * TK;                                                  \
    const int bi_ = (t) & 1;                                                   \
    tdm_load_2d(lds_offset_of(&la[bi_][0]), A + rowBase * DD + kb_, TK, 128,   \
                TK, 128, DD);                                                  \
    tdm_load_2d(lds_offset_of(&lb[bi_][0]), Bm + colBase * DD + kb_, TK, 128,  \
                TK, 128, DD);                                                  \
  } while (0)

  if (issuer) NS_ISSUE(0);
  for (int t = 0; t < NT; ++t) {
    if (issuer) {
      if (t + 1 < NT) {
        NS_ISSUE(t + 1);
        __builtin_amdgcn_s_wait_tensorcnt(2);  // tile t landed, t+1 in flight
      } else {
        __builtin_amdgcn_s_wait_tensorcnt(0);
      }
    }
    __syncthreads();
    mma_tile(&la[t & 1][0], &lb[t & 1][0], m0, n0, acc);
    __syncthreads();
  }
#undef NS_ISSUE

  const int lane = tid & 31, l15 = lane & 15, rh = (lane >> 4) * 8;
#pragma unroll
  for (int i = 0; i < 4; ++i)
#pragma unroll
    for (int j = 0; j < 2; ++j)
#pragma unroll
      for (int r = 0; r < 8; ++r) {
        const int row = rowBase + m0 + i * 16 + rh + r;
        const int col = colBase + n0 + j * 16 + l15;
        float v = acc[i][j][r] * scale;
        if (row == col) v += diag;
        C[row * DD + col] = v;
      }
}

// ---------------------------------------------------------------------------
// U[b, k(i,j)] = Y[b,i,j] * sqrt(normA[b]) for j >= i.
__global__ __launch_bounds__(256) void extract_kernel(const float* __restrict__ Y,
                                                      const float* __restrict__ norm,
                                                      float* __restrict__ U) {
  const int i = blockIdx.x, b = blockIdx.y, j = threadIdx.x;
  if (j < i) return;
  const int base = i * DD - (i * (i - 1)) / 2;
  U[(size_t)b * COVO + base + (j - i)] =
      Y[(size_t)b * DD * DD + i * DD + j] * sqrtf(norm[b]);
}

// ---------------------------------------------------------------------------
// out = U @ W^T + bias.  A=[batch][k], B=W=[class][k]; class clip via
// tensor_dim1 (OOB rows -> zeros), stores guarded.
__global__ __launch_bounds__(256) void fc_gemm(const float* __restrict__ U,
                                               const float* __restrict__ W,
                                               const float* __restrict__ bias,
                                               float* __restrict__ out) {
  __shared__ float la[2][128 * PT];
  __shared__ float lb[2][128 * PT];
  const int rowBase = (blockIdx.x & 1) * 128;
  const int colBase = (blockIdx.x >> 1) * 128;
  const int tid = threadIdx.x;
  const int wave = tid >> 5;
  const int m0 = (wave >> 2) * 64;
  const int n0 = (wave & 3) * 32;
  v8f acc[4][2];
#pragma unroll
  for (int i = 0; i < 4; ++i)
#pragma unroll
    for (int j = 0; j < 2; ++j) acc[i][j] = (v8f){0.f};

  const bool issuer = (wave == 0);
  const unsigned nvalid = (unsigned)(CLS - colBase);
  const int NT = COVO / TK;  // 2056 k-tiles
#define FC_ISSUE(t)                                                            \
  do {                                                                         \
    const int kb_ = (t) * TK;                                                  \
    const int bi_ = (t) & 1;                                                   \
    tdm_load_2d(lds_offset_of(&la[bi_][0]),                                    \
                U + (size_t)rowBase * COVO + kb_, TK, 128, TK, 128, COVO);     \
    tdm_load_2d(lds_offset_of(&lb[bi_][0]),                                    \
                W + (size_t)colBase * COVO + kb_, TK, 128, TK, nvalid, COVO);  \
  } while (0)

  if (issuer) FC_ISSUE(0);
  for (int t = 0; t < NT; ++t) {
    if (issuer) {
      if (t + 1 < NT) {
        FC_ISSUE(t + 1);
        __builtin_amdgcn_s_wait_tensorcnt(2);
      } else {
        __builtin_amdgcn_s_wait_tensorcnt(0);
      }
    }
    __syncthreads();
    mma_tile(&la[t & 1][0], &lb[t & 1][0], m0, n0, acc);
    __syncthreads();
  }
#undef FC_ISSUE

  const int lane = tid & 31, l15 = lane & 15, rh = (lane >> 4) * 8;
#pragma unroll
  for (int i = 0; i < 4; ++i)
#pragma unroll
    for (int j = 0; j < 2; ++j)
#pragma unroll
      for (int r = 0; r < 8; ++r) {
        const int row = rowBase + m0 + i * 16 + rh + r;
        const int col = colBase + n0 + j * 16 + l15;
        if (col < CLS) out[row * CLS + col] = acc[i][j][r] + bias[col];
      }
}

// ---------------------------------------------------------------------------
extern "C" void kernel_launch(void* const* d_in, const int* in_sizes, int n_in,
                              void* d_out, int out_size, void* d_ws, size_t ws_size,
                              hipStream_t stream) {
  (void)in_sizes; (void)n_in; (void)out_size; (void)ws_size;
  const float* x = (const float*)d_in[0];
  const float* fc_w = (const float*)d_in[1];
  const float* fc_b = (const float*)d_in[2];
  float* out = (float*)d_out;

  float* ws = (float*)d_ws;
  const size_t MAT = (size_t)BB * DD * DD;
  float* norm = ws;           // 256
  float* buf0 = ws + 512;     // 4 rotating 64MB matrix buffers
  float* buf1 = buf0 + MAT;
  float* buf2 = buf1 + MAT;
  float* buf3 = buf2 + MAT;

  // Ac lives in buf3 (49MB); consumed by cov_gemm before NS ever writes buf3.
  float* Acbuf = buf3;
  center_kernel<<<8192, 256, 0, stream>>>(x, Acbuf);
  cov_gemm<<<dim3(4, BB), 256, 0, stream>>>(Acbuf, buf0);
  norm_kernel<<<BB, 256, 0, stream>>>(buf0, norm);
  prep_kernel<<<65536, 256, 0, stream>>>(buf0, buf1, norm);

  float* Y = buf0;
  float* Z = buf1;
  float* t1 = buf2;
  float* t2 = buf3;
  for (int it = 0; it < 10; ++it) {
    float* T = t1;
    ns_gemm<<<dim3(4, BB), 256, 0, stream>>>(Z, Y, T, -0.5f, 1.5f);
    ns_gemm<<<dim3(4, BB), 256, 0, stream>>>(T, Z, t2, 1.0f, 0.0f);
    ns_gemm<<<dim3(4, BB), 256, 0, stream>>>(Y, T, Z, 1.0f, 0.0f);
    float* nY = Z;
    float* nZ = t2;
    t1 = Y;
    t2 = T;
    Y = nY;
    Z = nZ;
  }

  float* U = t1;  // freed 64MB buffer (needs 33.7MB)
  extract_kernel<<<dim3(DD, BB), 256, 0, stream>>>(Y, norm, U);
  fc_gemm<<<16, 256, 0, stream>>>(U, fc_w, fc_b, out);
}